// ConvAttention_46179488366772
// MI455X (gfx1250) — compile-verified
//
#include <hip/hip_runtime.h>

// Problem constants (from reference): x is (B,C,H,W,T) fp32
#define B_ 8
#define C_ 128
#define H_ 64
#define W_ 64
#define T_ 10
// x strides in floats: c-stride = H*W*T = 40960, h-stride = W*T = 640, w-stride = T = 10

typedef __attribute__((ext_vector_type(2))) float v2f;
typedef __attribute__((ext_vector_type(8))) float v8f;
typedef __attribute__((ext_vector_type(2))) int v2i;

// Workspace layout (floats):
//   [0, 3200)          WKeff[j][c']  (j = dy*5+dx in 0..24, c' in 0..127)
//   [4096, 4096+327680) p[b][h][w][s] attention weights (softmax over s)

#if defined(__has_builtin)
#if __has_builtin(__builtin_amdgcn_global_load_async_to_lds_b64) && \
    __has_builtin(__builtin_amdgcn_s_wait_asynccnt)
#define ASYNC_LDS 1
#endif
#endif

#if defined(ASYNC_LDS)
typedef __attribute__((address_space(1))) v2i GV2;  // global (AS1) v2i
typedef __attribute__((address_space(3))) v2i LV2;  // LDS (AS3) v2i
#endif

// ---------------------------------------------------------------------------
// K0: WKeff[j][c'] = sum_ck W2[0, 128+ck, j] * W1[128+ck, c']
// (bias terms of fK are constant over s and cancel in softmax -> skipped)
// ---------------------------------------------------------------------------
__global__ __launch_bounds__(256) void k0_weights(const float* __restrict__ W1,
                                                  const float* __restrict__ W2,
                                                  float* __restrict__ ws) {
  for (int i = threadIdx.x; i < 25 * 128; i += 256) {
    int j = i >> 7;
    int cp = i & 127;
    float acc = 0.f;
    for (int ck = 0; ck < 128; ++ck)
      acc += W2[(128 + ck) * 25 + j] * W1[(128 + ck) * 128 + cp];
    ws[i] = acc;
  }
}

// ---------------------------------------------------------------------------
// K1: fK[b,y,x,s] = sum_{j,c'} WKeff[j,c'] * x[b,c',y+dy-2,x+dx-2,s]  (zero pad)
//     p[b,y,x,s]  = softmax_s fK
// 16x16 spatial tile per block, channels in chunks of 4, LDS halo tile.
// ---------------------------------------------------------------------------
#define NC 4
#define TS 16
#define TP 20  // TS + 2*2 halo
__global__ __launch_bounds__(256) void k1_attn(const float* __restrict__ x,
                                               float* __restrict__ ws) {
  __shared__ float2 tile[NC * TP * TP * (T_ / 2)];  // 8000 float2 = 62.5 KB
  const float* WKeff = ws;
  float* p = ws + 4096;

  int blk = blockIdx.x;          // 0..127 : 8 b * 16 tiles
  int b = blk >> 4;
  int tb = blk & 15;
  int y0 = (tb >> 2) * TS;
  int x0 = (tb & 3) * TS;
  int tid = threadIdx.x;
  int ty = tid >> 4, tx = tid & 15;

  const float* xb = x + (size_t)b * (C_ * H_ * W_ * T_);
  float2 acc[5] = {};

  for (int c0 = 0; c0 < C_; c0 += NC) {
    __syncthreads();  // previous chunk's compute done before overwriting LDS
    for (int i = tid; i < NC * TP * TP * 5; i += 256) {
      int c = i / 2000;
      int r = i - c * 2000;
      int yy = r / 100;
      int r2 = r - yy * 100;
      int xx = r2 / 5;
      int sp = r2 - xx * 5;
      int gy = y0 - 2 + yy;
      int gx = x0 - 2 + xx;
      if ((unsigned)gy < (unsigned)H_ && (unsigned)gx < (unsigned)W_) {
        const float* src =
            xb + ((size_t)(c0 + c) * H_ + gy) * (W_ * T_) + gx * T_ + sp * 2;
#if defined(ASYNC_LDS)
        __builtin_amdgcn_global_load_async_to_lds_b64((GV2*)src, (LV2*)&tile[i], 0, 0);
#else
        tile[i] = *(const float2*)src;
#endif
      } else {
        tile[i] = make_float2(0.f, 0.f);
      }
    }
#if defined(ASYNC_LDS)
    __builtin_amdgcn_s_wait_asynccnt(0);
#endif
    __syncthreads();

    for (int c = 0; c < NC; ++c) {
      const float* wk = WKeff + (c0 + c);
#pragma unroll
      for (int dy = 0; dy < 5; ++dy) {
#pragma unroll
        for (int dx = 0; dx < 5; ++dx) {
          float wgt = wk[(dy * 5 + dx) * 128];  // wave-uniform -> scalar load
          const float2* src = &tile[(((c * TP + (ty + dy)) * TP) + (tx + dx)) * 5];
#pragma unroll
          for (int sp = 0; sp < 5; ++sp) {  // float2 math -> v_pk_fma_f32
            acc[sp].x += wgt * src[sp].x;
            acc[sp].y += wgt * src[sp].y;
          }
        }
      }
    }
  }

  // softmax over the 10 s values
  float f[T_];
#pragma unroll
  for (int sp = 0; sp < 5; ++sp) { f[2 * sp] = acc[sp].x; f[2 * sp + 1] = acc[sp].y; }
  float m = f[0];
#pragma unroll
  for (int s = 1; s < T_; ++s) m = fmaxf(m, f[s]);
  float sum = 0.f;
#pragma unroll
  for (int s = 0; s < T_; ++s) { f[s] = __expf(f[s] - m); sum += f[s]; }
  float inv = __frcp_rn(sum);
  float* pp = p + ((size_t)(b * H_ + (y0 + ty)) * W_ + (x0 + tx)) * T_;
#pragma unroll
  for (int sp = 0; sp < 5; ++sp)
    *(float2*)(pp + 2 * sp) = make_float2(f[2 * sp] * inv, f[2 * sp + 1] * inv);
}

// ---------------------------------------------------------------------------
// K2: xbar[c'] = sum_s p[s]*x[c',s] ;  out[c,t] = W1_V[c,:]·xbar + b1_V[c]
// One wave per 16-site group (16 consecutive w in one row).
// GEMM tile: M=16 c_out, N=16 sites, K=128 via V_WMMA_F32_16X16X4_F32 (exact fp32).
// Output replicated over t (softmax over s is t-independent).
// ---------------------------------------------------------------------------
__global__ __launch_bounds__(128) void k2_out(const float* __restrict__ x,
                                              const float* __restrict__ W1,
                                              const float* __restrict__ b1,
                                              const float* __restrict__ ws,
                                              float* __restrict__ out) {
  __shared__ float xbar[4][C_ * 16];  // 32 KB, one 128x16 panel per wave
  const float* p = ws + 4096;
  const float* W1v = W1 + 256 * 128;  // V block of W1
  const float* b1v = b1 + 256;

  int wave = threadIdx.x >> 5;
  int lane = threadIdx.x & 31;
  int g = blockIdx.x * 4 + wave;  // 0..2047 site groups
  int b = g >> 8;
  int rem = g & 255;
  int h = rem >> 2;
  int w0 = (rem & 3) << 4;

  const float* xbh = x + ((size_t)b * C_ * H_ + h) * (W_ * T_);  // x[b,0,h,0,0]
  int n = lane & 15;       // site within group
  int half = lane >> 4;    // 0/1

  float ps[T_];
  const float* pp = p + ((size_t)(b * H_ + h) * W_ + (w0 + n)) * T_;
#pragma unroll
  for (int s = 0; s < T_; ++s) ps[s] = pp[s];

  float* sh = xbar[wave];
  for (int i = 0; i < 64; ++i) {  // c' = half + 2*i covers 0..127 across the wave
    int c = half + 2 * i;
    const float* xs = xbh + (size_t)c * (H_ * W_ * T_) + (w0 + n) * T_;
    float a = 0.f;
#pragma unroll
    for (int s = 0; s < T_; ++s) a += ps[s] * xs[s];
    sh[c * 16 + n] = a;
  }
  __syncthreads();

  int m = lane & 15;               // A-matrix row (c_out within block)
  int koff = (lane >> 4) << 1;     // lanes 16..31 hold K+2,K+3
  size_t outbase = ((size_t)b * C_ * H_ + h) * (W_ * T_);

  for (int cb = 0; cb < 8; ++cb) {
    v8f acc = {};
#pragma unroll 4
    for (int kk = 0; kk < 32; ++kk) {
      int kb = kk * 4;
      const float* wrow = W1v + (size_t)(cb * 16 + m) * 128 + kb + koff;
      v2f a;  a[0] = wrow[0];               a[1] = wrow[1];
      v2f bb; bb[0] = sh[(kb + koff) * 16 + n]; bb[1] = sh[(kb + koff + 1) * 16 + n];
      acc = __builtin_amdgcn_wmma_f32_16x16x4_f32(
          /*neg_a=*/false, a, /*neg_b=*/false, bb,
          /*c_mod=*/(short)0, acc, /*reuse_a=*/false, /*reuse_b=*/false);
    }
#pragma unroll
    for (int r = 0; r < 8; ++r) {
      int c = cb * 16 + r + ((lane >= 16) ? 8 : 0);
      float v = acc[r] + b1v[c];
      float* o = out + outbase + (size_t)c * (H_ * W_ * T_) + (w0 + n) * T_;
      float2 v2 = make_float2(v, v);  // out is constant over t
#pragma unroll
      for (int q = 0; q < 5; ++q) *(float2*)(o + 2 * q) = v2;  // 40B-aligned b64 stores
    }
  }
}

extern "C" void kernel_launch(void* const* d_in, const int* in_sizes, int n_in,
                              void* d_out, int out_size, void* d_ws, size_t ws_size,
                              hipStream_t stream) {
  const float* x  = (const float*)d_in[0];
  const float* W1 = (const float*)d_in[1];
  const float* b1 = (const float*)d_in[2];
  const float* W2 = (const float*)d_in[3];
  float* ws = (float*)d_ws;
  float* out = (float*)d_out;

  k0_weights<<<1, 256, 0, stream>>>(W1, W2, ws);
  k1_attn<<<128, 256, 0, stream>>>(x, ws);
  k2_out<<<512, 128, 0, stream>>>(x, W1, b1, ws, out);
}